// GAT_78881369359026
// MI455X (gfx1250) — compile-verified
//
#include <hip/hip_runtime.h>
#include <hip/hip_bf16.h>
#include <math.h>

#define NEG_SLOPE 0.2f

typedef float v2f __attribute__((ext_vector_type(2)));
typedef float v8f __attribute__((ext_vector_type(8)));

// ---------------------------------------------------------------------------
// Async global -> LDS copy (CDNA5 GLOBAL_LOAD_ASYNC_TO_LDS_B32, ASYNCcnt).
// Probe round 2 showed the builtin exists and its first parameter is
// 'address_space(1) int*' (non-const); second is the LDS (AS3) pointer.
// ---------------------------------------------------------------------------
typedef __attribute__((address_space(1))) int as1_int;
typedef __attribute__((address_space(3))) int as3_int;

__device__ __forceinline__ void async_cp_f32(const float* g, float* l) {
#if __has_builtin(__builtin_amdgcn_global_load_async_to_lds_b32)
  as1_int* gp = (as1_int*)const_cast<float*>(g);
  as3_int* lp = (as3_int*)l;
  __builtin_amdgcn_global_load_async_to_lds_b32(gp, lp, 0, 0);
#else
  *l = *g;  // synchronous fallback
#endif
}

__device__ __forceinline__ void async_wait0() {
#if __has_builtin(__builtin_amdgcn_s_wait_asynccnt)
  __builtin_amdgcn_s_wait_asynccnt(0);
#elif __has_builtin(__builtin_amdgcn_global_load_async_to_lds_b32)
  asm volatile("s_wait_asynccnt 0x0" ::: "memory");
#endif
}

// ---------------------------------------------------------------------------
// WMMA f32 GEMM: C[M,Nn] = A[M,K] x B[K,Nn], row-major.
// Requires M%16==0, Nn%16==0, K%4==0.
//
// Block = 256 threads = 8 waves. Each block owns a 128-row band x one 16-col
// tile. The K x 16 panel of B is staged ONCE per block into LDS with async
// global->LDS copies (s_wait_asynccnt + barrier fence), then all 8 waves run
// the V_WMMA_F32_16X16X4_F32 K-loop with A from global (L2-resident) and B
// from LDS (conflict-free ds reads).
//
// Register layouts per CDNA5 ISA 7.12.2:
//   A (16x4, MxK):  lane l: M = l%16; vgpr v: K = v + 2*(l/16)  -> float2 load
//   B (4x16, KxN):  lane l: N = l%16; vgpr v: K = v + 2*(l/16)
//   C/D (16x16):    lane l: N = l%16; vgpr v: M = v + 8*(l/16)
// ---------------------------------------------------------------------------
__global__ void __launch_bounds__(256)
gemm_wmma_f32(const float* __restrict__ A, const float* __restrict__ B,
              float* __restrict__ C, int M, int Nn, int K) {
  extern __shared__ float Blds[];  // K x 16 floats (row-major, n fastest)

  const int tid    = threadIdx.x;
  const int lane   = tid & 31;
  const int wave   = tid >> 5;
  const int tilesM = M >> 4;
  const int tilesN = Nn >> 4;
  const int band   = blockIdx.x / tilesN;   // 8 M-tiles per band
  const int nt     = blockIdx.x % tilesN;
  const int tn     = nt << 4;

  // ---- stage B[K x 16] panel into LDS (all threads participate) ----
  const int panel = K << 4;  // K*16 elements
  for (int idx = tid; idx < panel; idx += 256) {
    const int k = idx >> 4, n = idx & 15;
    async_cp_f32(B + (size_t)k * Nn + tn + n, Blds + idx);
  }
  async_wait0();
  __syncthreads();

  // ---- per-wave 16x16 tile compute ----
  const int tmTile = band * 8 + wave;
  if (tmTile < tilesM) {
    const int tm   = tmTile << 4;
    const int half = lane >> 4;   // 0: lanes 0-15, 1: lanes 16-31
    const int l16  = lane & 15;

    v8f acc = {};
    const float* Arow = A + (size_t)(tm + l16) * K;  // this lane's A row
    for (int kk = 0; kk < K; kk += 4) {
      const int k0 = kk + (half << 1);
      v2f a, b;
      a.x = Arow[k0];
      a.y = Arow[k0 + 1];
      b.x = Blds[(k0 << 4) + l16];         // B[k0][l16]
      b.y = Blds[((k0 + 1) << 4) + l16];   // B[k0+1][l16]
      // 8 args: (neg_a, A, neg_b, B, c_mod, C, reuse_a, reuse_b)
      acc = __builtin_amdgcn_wmma_f32_16x16x4_f32(
          false, a, false, b, (short)0, acc, false, false);
    }
    float* Cp = C + (size_t)(tm + (half << 3)) * Nn + tn + l16;
#pragma unroll
    for (int v = 0; v < 8; ++v) Cp[(size_t)v * Nn] = acc[v];
  }
}

// Small GEMM for the skinny 16->2 layer (640k MACs total; not worth a tile).
__global__ void gemm_small(const float* __restrict__ A, const float* __restrict__ B,
                           float* __restrict__ C, int M, int Nn, int K) {
  int i = blockIdx.x * blockDim.x + threadIdx.x;
  if (i >= M * Nn) return;
  int m = i / Nn, n = i % Nn;
  float s = 0.f;
  for (int k = 0; k < K; ++k) s += A[(size_t)m * K + k] * B[(size_t)k * Nn + n];
  C[i] = s;
}

// Per-node attention scalars: as[n] = h[n].a_src, ad[n] = h[n].a_dst.
// One wave32 per node, butterfly reduction.
__global__ void alpha_kernel(const float* __restrict__ h,
                             const float* __restrict__ a_src,
                             const float* __restrict__ a_dst,
                             float* __restrict__ as_, float* __restrict__ ad_,
                             int N, int F) {
  int wave = (blockIdx.x * blockDim.x + threadIdx.x) >> 5;
  int lane = threadIdx.x & 31;
  if (wave >= N) return;
  const float* row = h + (size_t)wave * F;
  float s = 0.f, d = 0.f;
  for (int f = lane; f < F; f += 32) {
    float v = row[f];
    s += v * a_src[f];
    d += v * a_dst[f];
  }
#pragma unroll
  for (int off = 16; off; off >>= 1) {
    s += __shfl_down(s, off, 32);
    d += __shfl_down(d, off, 32);
  }
  if (lane == 0) { as_[wave] = s; ad_[wave] = d; }
}

// Order-preserving float<->uint encoding so unsigned atomicMax == float max.
// Key 0 decodes below -inf, so memset(0) is a safe init (self-loops guarantee
// every destination gets at least one update).
__device__ __forceinline__ unsigned f2ord(float f) {
  unsigned u = __float_as_uint(f);
  return (u & 0x80000000u) ? ~u : (u | 0x80000000u);
}
__device__ __forceinline__ float ord2f(unsigned k) {
  return __uint_as_float((k & 0x80000000u) ? (k ^ 0x80000000u) : ~k);
}

// Pass 1: logits = leaky_relu(as[s] + ad[d]); per-destination running max.
__global__ void edge_logit_max(const int* __restrict__ src, const int* __restrict__ dst,
                               const float* __restrict__ as_, const float* __restrict__ ad_,
                               float* __restrict__ logits, unsigned* __restrict__ maxb,
                               int E, int N) {
  int i = blockIdx.x * blockDim.x + threadIdx.x;
  if (i >= E + N) return;
  int s = (i < E) ? src[i] : (i - E);   // tail = self loops
  int d = (i < E) ? dst[i] : (i - E);
  float x = as_[s] + ad_[d];
  float l = (x > 0.f) ? x : NEG_SLOPE * x;
  logits[i] = l;
  atomicMax(maxb + d, f2ord(l));
}

// Pass 2: e = exp(logit - max[d]); per-destination sum.
__global__ void edge_expsum(const int* __restrict__ dst,
                            const float* __restrict__ logits,
                            const unsigned* __restrict__ maxb,
                            float* __restrict__ ebuf, float* __restrict__ sumb,
                            int E, int N) {
  int i = blockIdx.x * blockDim.x + threadIdx.x;
  if (i >= E + N) return;
  int d = (i < E) ? dst[i] : (i - E);
  float e = __expf(logits[i] - ord2f(maxb[d]));
  ebuf[i] = e;
  atomicAdd(sumb + d, e);
}

// Pass 3: out[d] += h[s] * (e / sum[d]).  LPE lanes cooperate per edge so the
// feature-dim atomics are coalesced; atomics resolve in L2 (out fits easily
// in the 192MB L2).
template <int LPE>
__global__ void edge_aggregate(const int* __restrict__ src, const int* __restrict__ dst,
                               const float* __restrict__ h,
                               const float* __restrict__ ebuf,
                               const float* __restrict__ sumb,
                               float* __restrict__ out, int E, int N, int F) {
  int gid = blockIdx.x * blockDim.x + threadIdx.x;
  int i = gid / LPE;
  int sub = gid % LPE;
  if (i >= E + N) return;
  int s = (i < E) ? src[i] : (i - E);
  int d = (i < E) ? dst[i] : (i - E);
  float w = ebuf[i] / (sumb[d] + 1e-16f);
  const float* hs = h + (size_t)s * F;
  float* od = out + (size_t)d * F;
  for (int f = sub; f < F; f += LPE) atomicAdd(od + f, hs[f] * w);
}

// Bias + activation. mode: 0 = none, 1 = relu, 2 = sigmoid.
__global__ void bias_act(float* __restrict__ x, const float* __restrict__ b,
                         int N, int F, int mode) {
  int i = blockIdx.x * blockDim.x + threadIdx.x;
  if (i >= N * F) return;
  float v = x[i] + b[i % F];
  if (mode == 1) v = fmaxf(v, 0.f);
  else if (mode == 2) v = 1.f / (1.f + __expf(-v));
  x[i] = v;
}

extern "C" void kernel_launch(void* const* d_in, const int* in_sizes, int n_in,
                              void* d_out, int out_size, void* d_ws, size_t ws_size,
                              hipStream_t stream) {
  const float* x  = (const float*)d_in[0];
  const int*   ei = (const int*)d_in[1];
  const float* W1 = (const float*)d_in[2];
  const float* aS1 = (const float*)d_in[3];
  const float* aD1 = (const float*)d_in[4];
  const float* b1 = (const float*)d_in[5];
  const float* W2 = (const float*)d_in[6];
  const float* aS2 = (const float*)d_in[7];
  const float* aD2 = (const float*)d_in[8];
  const float* b2 = (const float*)d_in[9];
  const float* W3 = (const float*)d_in[10];
  const float* aS3 = (const float*)d_in[11];
  const float* aD3 = (const float*)d_in[12];
  const float* b3 = (const float*)d_in[13];

  const int H1  = in_sizes[3];                 // 256
  const int Fin = in_sizes[2] / H1;            // 256
  const int N   = in_sizes[0] / Fin;           // 20000
  const int E   = in_sizes[1] / 2;             // 640000
  const int H2  = in_sizes[7];                 // 16
  const int OUT = in_sizes[11];                // 2
  const int T   = E + N;                       // edges incl. self loops

  const int* src = ei;
  const int* dst = ei + E;

  // ---- workspace carve-out (256B aligned slabs); total ~49 MB ----
  char* wp = (char*)d_ws;
  auto alloc = [&](size_t bytes) -> void* {
    void* p = wp;
    wp += (bytes + 255) & ~(size_t)255;
    return p;
  };
  float*    h1   = (float*)alloc((size_t)N * H1 * 4);
  float*    o1   = (float*)alloc((size_t)N * H1 * 4);
  float*    h2   = (float*)alloc((size_t)N * H2 * 4);
  float*    o2   = (float*)alloc((size_t)N * H2 * 4);
  float*    h3   = (float*)alloc((size_t)N * OUT * 4);
  float*    as_  = (float*)alloc((size_t)N * 4);
  float*    ad_  = (float*)alloc((size_t)N * 4);
  float*    lg   = (float*)alloc((size_t)T * 4);
  float*    eb   = (float*)alloc((size_t)T * 4);
  unsigned* mx   = (unsigned*)alloc((size_t)N * 4);
  float*    sm   = (float*)alloc((size_t)N * 4);

  const int TPB = 256;
  auto run_layer = [&](const float* in, int Fi, int Fo, const float* W,
                       const float* aSv, const float* aDv, const float* bv,
                       float* hbuf, float* obuf, int actMode, bool useWmma) {
    if (useWmma) {
      int tilesM = N >> 4, tilesN = Fo >> 4;
      int bands = (tilesM + 7) / 8;                  // 8 M-tiles (waves) / block
      size_t shmem = (size_t)Fi * 16 * sizeof(float);  // B panel (K x 16)
      gemm_wmma_f32<<<bands * tilesN, TPB, shmem, stream>>>(in, W, hbuf, N, Fo, Fi);
    } else {
      int t = N * Fo;
      gemm_small<<<(t + TPB - 1) / TPB, TPB, 0, stream>>>(in, W, hbuf, N, Fo, Fi);
    }
    (void)hipMemsetAsync(obuf, 0, (size_t)N * Fo * 4, stream);
    (void)hipMemsetAsync(mx, 0, (size_t)N * 4, stream);
    (void)hipMemsetAsync(sm, 0, (size_t)N * 4, stream);

    alpha_kernel<<<(N * 32 + TPB - 1) / TPB, TPB, 0, stream>>>(hbuf, aSv, aDv, as_, ad_, N, Fo);
    edge_logit_max<<<(T + TPB - 1) / TPB, TPB, 0, stream>>>(src, dst, as_, ad_, lg, mx, E, N);
    edge_expsum<<<(T + TPB - 1) / TPB, TPB, 0, stream>>>(dst, lg, mx, eb, sm, E, N);

    if (Fo >= 32) {
      long th = (long)T * 32;
      edge_aggregate<32><<<(unsigned)((th + TPB - 1) / TPB), TPB, 0, stream>>>(src, dst, hbuf, eb, sm, obuf, E, N, Fo);
    } else if (Fo >= 16) {
      long th = (long)T * 16;
      edge_aggregate<16><<<(unsigned)((th + TPB - 1) / TPB), TPB, 0, stream>>>(src, dst, hbuf, eb, sm, obuf, E, N, Fo);
    } else {
      long th = (long)T * 2;
      edge_aggregate<2><<<(unsigned)((th + TPB - 1) / TPB), TPB, 0, stream>>>(src, dst, hbuf, eb, sm, obuf, E, N, Fo);
    }
    bias_act<<<((long)N * Fo + TPB - 1) / TPB, TPB, 0, stream>>>(obuf, bv, N, Fo, actMode);
  };

  // Layer 1: 256 -> 256 (WMMA + async LDS staging), relu
  run_layer(x, Fin, H1, W1, aS1, aD1, b1, h1, o1, /*relu*/ 1, true);
  // Layer 2: 256 -> 16 (WMMA + async LDS staging), relu
  run_layer(o1, H1, H2, W2, aS2, aD2, b2, h2, o2, /*relu*/ 1, true);
  // Layer 3: 16 -> 2 (plain), sigmoid, aggregate straight into d_out
  run_layer(o2, H2, OUT, W3, aS3, aD3, b3, h3, (float*)d_out, /*sigmoid*/ 2, false);
}